// rnnbptt_45329084842011
// MI455X (gfx1250) — compile-verified
//
#include <hip/hip_runtime.h>
#include <math.h>
#include <stdint.h>

// ---------------- model dimensions (match reference setup_inputs) ----------
#define T_DIM   512
#define B_DIM   128
#define MD_DIM  31
#define H_DIM   512
#define E_DIM   256
#define S_DIM   128
#define G3_DIM  1536                       // 3*H
#define NROWS   (T_DIM * B_DIM)            // 65536 flattened (t,b) rows
#define SIGMA_MIN 0.01f
#define LOG2PI_F  1.8378770664093453f
#define GAMMA_F   1.0f

typedef float v2f __attribute__((ext_vector_type(2)));
typedef float v8f __attribute__((ext_vector_type(8)));

// ---------------- fp32 WMMA: D = A(16x4) * B(4x16) + C (16x16) -------------
__device__ __forceinline__ v8f wmma4(v2f a, v2f b, v8f c) {
    // 8 args: (neg_a, A, neg_b, B, c_mod, C, reuse_a, reuse_b)
    return __builtin_amdgcn_wmma_f32_16x16x4_f32(
        false, a, false, b, (short)0, c, false, false);
}

// ---------------- LDS staging geometry for the GRU step --------------------
// W_hh strips are stored with TDM padding: 2 dwords inserted every 256 dwords
// => element (r,k) of a 16x512 strip lives at dword offset r*516 + k + 2*(k>>8).
// Lane lm's fragment base is lm*516 dwords => (lm*4) mod 64 banks: the 16
// lanes of a b64 LDS fragment read land on disjoint bank groups (no conflict).
#define LDS_ROW   516                  // 512 + 2x2 pad dwords
#define LDS_STRIP (16 * LDS_ROW)       // dwords per gate strip (33,024 B)
#define GRU_SMEM_BYTES (3 * LDS_STRIP * 4)

#if __has_builtin(__builtin_amdgcn_tensor_load_to_lds)
#define HAVE_TDM 1
typedef unsigned int u32x4 __attribute__((ext_vector_type(4)));
typedef int          i32x4 __attribute__((ext_vector_type(4)));
typedef int          i32x8 __attribute__((ext_vector_type(8)));

// Issue one TDM load of a 16x512 f32 tile of W_hh (row-major, ld=512) into
// LDS at lds_byte_addr, with 2-dword padding every 256 dwords (bank de-skew).
// D# layout per CDNA5 ISA.
__device__ __forceinline__ void tdm_load_whh_strip(unsigned lds_byte_addr,
                                                   const float* gsrc)
{
    const unsigned long long ga = (unsigned long long)(uintptr_t)gsrc;
    u32x4 g0;
    g0[0] = 1u;                                    // count=1 (valid user D#)
    g0[1] = lds_byte_addr;                         // bits 63:32  lds_addr
    g0[2] = (unsigned)(ga & 0xFFFFFFFFu);          // global_addr[31:0]
    g0[3] = (unsigned)((ga >> 32) & 0x01FFFFFFu)   // global_addr[56:32]
          | (2u << 30);                            // type = 2 ("image")
    i32x8 g1;
    g1[0] = (2 << 16)                              // data_size = 4 bytes
          | (1 << 20)                              // pad_enable
          | (7 << 22)                              // pad_interval = 256 dwords
          | (1 << 25);                             // pad_amount   = 2 dwords
    g1[1] = (512 << 16);                           // tensor_dim0 = 512
    g1[2] = (1536 << 16);                          // tensor_dim1 = 1536
    g1[3] = (512 << 16);                           // tile_dim0 = 512
    g1[4] = 16;                                    // tile_dim1 = 16, tile_dim2 = 0
    g1[5] = 512;                                   // tensor_dim0_stride = 512
    g1[6] = 0;
    g1[7] = 0;
    i32x4 z4 = {0, 0, 0, 0};
#if __clang_major__ >= 23
    i32x8 z8 = {0, 0, 0, 0, 0, 0, 0, 0};
    __builtin_amdgcn_tensor_load_to_lds(g0, g1, z4, z4, z8, 0);
#else
    __builtin_amdgcn_tensor_load_to_lds(g0, g1, z4, z4, 0);
#endif
}
#else
#define HAVE_TDM 0
#endif

// =====================================================================
// Kernel 1: phi[n,e] = ReLU( x[n,:31] . W_embed[e,:31] + b_embed[e] )
//   65536 x 256, K = 31 (padded to 32 with zero guards)
// =====================================================================
__global__ __launch_bounds__(256) void k_embed(
    const float* __restrict__ X, const float* __restrict__ We,
    const float* __restrict__ be, float* __restrict__ phi)
{
    const int lane = threadIdx.x & 31;
    const int half = lane >> 4, lm = lane & 15;
    const int gw = blockIdx.x * 8 + (threadIdx.x >> 5);
    const int tn = gw >> 4;                 // row tile: 0..4095
    const int te = gw & 15;                 // col tile: 0..15
    const int row0 = tn * 16, col0 = te * 16;
    const float* __restrict__ arow = X  + (size_t)(row0 + lm) * MD_DIM;
    const float* __restrict__ brow = We + (size_t)(col0 + lm) * MD_DIM;

    v8f c = {};
#pragma unroll
    for (int k0 = 0; k0 < 32; k0 += 4) {
        const int kk = k0 + 2 * half;
        v2f a, b;
        a.x = (kk     < MD_DIM) ? arow[kk]     : 0.0f;
        a.y = (kk + 1 < MD_DIM) ? arow[kk + 1] : 0.0f;
        b.x = (kk     < MD_DIM) ? brow[kk]     : 0.0f;
        b.y = (kk + 1 < MD_DIM) ? brow[kk + 1] : 0.0f;
        c = wmma4(a, b, c);
    }
#pragma unroll
    for (int i = 0; i < 8; ++i) {
        const int r   = row0 + i + 8 * half;
        const int col = col0 + lm;
        float v = c[i] + be[col];
        phi[(size_t)r * E_DIM + col] = v > 0.0f ? v : 0.0f;
    }
}

// =====================================================================
// Kernel 2: xg[n,g] = phi[n,:256] . W_ih[g,:256] + b_ih[g]
//   65536 x 1536, K = 256
// =====================================================================
__global__ __launch_bounds__(256) void k_xgates(
    const float* __restrict__ phi, const float* __restrict__ Wih,
    const float* __restrict__ bih, float* __restrict__ xg)
{
    const int lane = threadIdx.x & 31;
    const int half = lane >> 4, lm = lane & 15;
    const int gw = blockIdx.x * 8 + (threadIdx.x >> 5);
    const int tn = gw / 96;                 // 0..4095
    const int tg = gw % 96;                 // 0..95
    const int row0 = tn * 16, col0 = tg * 16;
    const float* __restrict__ arow = phi + (size_t)(row0 + lm) * E_DIM;
    const float* __restrict__ brow = Wih + (size_t)(col0 + lm) * E_DIM;

    v8f c = {};
#pragma unroll 4
    for (int k0 = 0; k0 < E_DIM; k0 += 4) {
        const int kk = k0 + 2 * half;
        v2f a, b;
        a.x = arow[kk]; a.y = arow[kk + 1];
        b.x = brow[kk]; b.y = brow[kk + 1];
        c = wmma4(a, b, c);
    }
#pragma unroll
    for (int i = 0; i < 8; ++i) {
        const int r   = row0 + i + 8 * half;
        const int col = col0 + lm;
        xg[(size_t)r * G3_DIM + col] = c[i] + bih[col];
    }
}

// =====================================================================
// Kernel 3 (x512): one GRU step.
//   32 blocks x 256 threads. Block -> hidden-column tile j0 = 16*blockIdx;
//   wave w (0..7) -> batch tile b0 = 16*w. Wave 0 TDM-loads the three
//   16x512 W_hh gate strips (r,z,n) into padded LDS; all 8 waves reuse the
//   B-fragments from LDS while streaming their own A-fragments (hidden
//   state) from global/L2. Three f32 WMMA accumulators per wave share one
//   A-fragment per k-step.
// =====================================================================
__global__ __launch_bounds__(256) void k_gru_step(
    const float* __restrict__ Hin, float* __restrict__ Hout,
    const float* __restrict__ xg_t, const float* __restrict__ Whh,
    const float* __restrict__ bhh)
{
    extern __shared__ float smem[];               // 3 * LDS_STRIP dwords
    const int lane = threadIdx.x & 31;
    const int half = lane >> 4, lm = lane & 15;
    const int wv = threadIdx.x >> 5;              // 0..7  -> batch tile
    const int j0 = blockIdx.x * 16;               // hidden column base
    const int b0 = wv * 16;

#if HAVE_TDM
    if (wv == 0) {                                // one TDM issue per strip
        const unsigned lds0 = (unsigned)(uintptr_t)&smem[0];
        tdm_load_whh_strip(lds0,                       Whh + (size_t)(0 * H_DIM + j0) * H_DIM);
        tdm_load_whh_strip(lds0 + 1u * LDS_STRIP * 4u, Whh + (size_t)(1 * H_DIM + j0) * H_DIM);
        tdm_load_whh_strip(lds0 + 2u * LDS_STRIP * 4u, Whh + (size_t)(2 * H_DIM + j0) * H_DIM);
        __builtin_amdgcn_s_wait_tensorcnt(0);
    }
    __syncthreads();
#else
    // cooperative fallback: stage the strips with plain loads (same layout)
    for (int idx = threadIdx.x; idx < 16 * H_DIM; idx += 256) {
        const int r = idx >> 9, k = idx & 511;
        const int lo = r * LDS_ROW + k + 2 * (k >> 8);
        smem[0 * LDS_STRIP + lo] = Whh[(size_t)(0 * H_DIM + j0 + r) * H_DIM + k];
        smem[1 * LDS_STRIP + lo] = Whh[(size_t)(1 * H_DIM + j0 + r) * H_DIM + k];
        smem[2 * LDS_STRIP + lo] = Whh[(size_t)(2 * H_DIM + j0 + r) * H_DIM + k];
    }
    __syncthreads();
#endif

    const float* __restrict__ arow = Hin + (size_t)(b0 + lm) * H_DIM;
    // prefetch the x-gate row this wave's epilogue will consume
    __builtin_prefetch(xg_t + (size_t)(b0 + lm) * G3_DIM + j0, 0, 1);

    const int roff = lm * LDS_ROW;
    v8f cr = {}, cz = {}, cn = {};
#pragma unroll 4
    for (int k0 = 0; k0 < H_DIM; k0 += 4) {
        const int kk = k0 + 2 * half;
        const int lo = roff + kk + 2 * (kk >> 8);
        v2f a, br, bz, bn;
        a.x  = arow[kk];                  a.y  = arow[kk + 1];
        br.x = smem[0 * LDS_STRIP + lo];  br.y = smem[0 * LDS_STRIP + lo + 1];
        bz.x = smem[1 * LDS_STRIP + lo];  bz.y = smem[1 * LDS_STRIP + lo + 1];
        bn.x = smem[2 * LDS_STRIP + lo];  bn.y = smem[2 * LDS_STRIP + lo + 1];
        cr = wmma4(a, br, cr);
        cz = wmma4(a, bz, cz);
        cn = wmma4(a, bn, cn);
    }
#pragma unroll
    for (int i = 0; i < 8; ++i) {
        const int b = b0 + i + 8 * half;
        const int j = j0 + lm;
        const float* __restrict__ xr = xg_t + (size_t)b * G3_DIM;
        const float hr = cr[i] + bhh[j];
        const float hz = cz[i] + bhh[H_DIM + j];
        const float hn = cn[i] + bhh[2 * H_DIM + j];
        const float r = 1.0f / (1.0f + expf(-(xr[j] + hr)));
        const float z = 1.0f / (1.0f + expf(-(xr[H_DIM + j] + hz)));
        const float n = tanhf(xr[2 * H_DIM + j] + r * hn);
        const float hold = Hin[(size_t)b * H_DIM + j];
        Hout[(size_t)b * H_DIM + j] = (1.0f - z) * n + z * hold;
    }
}

// =====================================================================
// Kernel 4: he[n,s] = ReLU( hs[n,:512] . W_he[s,:512] + b_he[s] )
//   65536 x 128, K = 512   (hs == Hbuf rows 0..T*B-1, pre-update states)
// =====================================================================
__global__ __launch_bounds__(256) void k_he(
    const float* __restrict__ Hs, const float* __restrict__ Whe,
    const float* __restrict__ bhe, float* __restrict__ he)
{
    const int lane = threadIdx.x & 31;
    const int half = lane >> 4, lm = lane & 15;
    const int gw = blockIdx.x * 8 + (threadIdx.x >> 5);
    const int tn = gw >> 3;                 // 0..4095
    const int ts = gw & 7;                  // 0..7
    const int row0 = tn * 16, col0 = ts * 16;
    const float* __restrict__ arow = Hs  + (size_t)(row0 + lm) * H_DIM;
    const float* __restrict__ brow = Whe + (size_t)(col0 + lm) * H_DIM;

    v8f c = {};
#pragma unroll 4
    for (int k0 = 0; k0 < H_DIM; k0 += 4) {
        const int kk = k0 + 2 * half;
        v2f a, b;
        a.x = arow[kk]; a.y = arow[kk + 1];
        b.x = brow[kk]; b.y = brow[kk + 1];
        c = wmma4(a, b, c);
    }
#pragma unroll
    for (int i = 0; i < 8; ++i) {
        const int r   = row0 + i + 8 * half;
        const int col = col0 + lm;
        float v = c[i] + bhe[col];
        he[(size_t)r * S_DIM + col] = v > 0.0f ? v : 0.0f;
    }
}

// =====================================================================
// Kernel 5: per-row heads. One wave per row n = t*B + b.
// =====================================================================
__global__ __launch_bounds__(256) void k_head(
    const float* __restrict__ he, const float* __restrict__ X,
    const float* __restrict__ Tm, const float* __restrict__ mask,
    const float* __restrict__ Wmu, const float* __restrict__ bmu,
    const float* __restrict__ Wlv, const float* __restrict__ blv,
    const float* __restrict__ hinf, const float* __restrict__ ti_p,
    const float* __restrict__ bi_p, float* __restrict__ partial)
{
    const int lane = threadIdx.x & 31;
    const int n = blockIdx.x * 8 + (threadIdx.x >> 5);
    const float* __restrict__ hrow = he + (size_t)n * S_DIM;

    float term = 0.0f;
    if (lane < MD_DIM) {
        float mu = bmu[lane], lv = blv[lane];
        const float* __restrict__ wm = Wmu + lane * S_DIM;
        const float* __restrict__ wl = Wlv + lane * S_DIM;
#pragma unroll 8
        for (int s = 0; s < S_DIM; ++s) {
            const float h = hrow[s];
            mu = fmaf(h, wm[s], mu);
            lv = fmaf(h, wl[s], lv);
        }
        float sigma = expf(0.5f * lv);
        sigma = sigma > SIGMA_MIN ? sigma : SIGMA_MIN;
        const float d = (X[(size_t)n * MD_DIM + lane] - mu) / sigma;
        term = -0.5f * d * d - logf(sigma) - 0.5f * LOG2PI_F;
    }
    float ps = 0.0f;
#pragma unroll
    for (int s = lane; s < S_DIM; s += 32)
        ps = fmaf(hrow[s], hinf[s], ps);

#pragma unroll
    for (int off = 16; off > 0; off >>= 1) {
        term += __shfl_down(term, off, 32);
        ps   += __shfl_down(ps,   off, 32);
    }
    if (lane == 0) {
        const float marker_ll = term;
        const float past = ps;
        const float ti = ti_p[0], bi = bi_p[0];
        const float dt = Tm[(size_t)n * 2 + 1];
        const float term1 = past + ti * dt + bi;
        const float time_ll = term1 + (expf(past + bi) - expf(term1)) / ti;
        const int tstep = n / B_DIM;
        float out = 0.0f;
        if (tstep >= 1)
            out = -(GAMMA_F * time_ll + marker_ll) * mask[n];
        partial[n] = out;
    }
}

// =====================================================================
// Kernel 6: deterministic block tree-reduction (65536 -> 256 -> 1)
// =====================================================================
__global__ __launch_bounds__(256) void k_reduce(
    const float* __restrict__ in, float* __restrict__ out)
{
    __shared__ float s[256];
    const int i = blockIdx.x * 256 + threadIdx.x;
    s[threadIdx.x] = in[i];
    __syncthreads();
#pragma unroll
    for (int off = 128; off > 0; off >>= 1) {
        if (threadIdx.x < off) s[threadIdx.x] += s[threadIdx.x + off];
        __syncthreads();
    }
    if (threadIdx.x == 0) out[blockIdx.x] = s[0];
}

// =====================================================================
extern "C" void kernel_launch(void* const* d_in, const int* in_sizes, int n_in,
                              void* d_out, int out_size, void* d_ws, size_t ws_size,
                              hipStream_t stream)
{
    const float* x       = (const float*)d_in[0];
    const float* t       = (const float*)d_in[1];
    const float* mask    = (const float*)d_in[2];
    const float* W_embed = (const float*)d_in[3];
    const float* b_embed = (const float*)d_in[4];
    const float* W_ih    = (const float*)d_in[5];
    const float* b_ih    = (const float*)d_in[6];
    const float* W_hh    = (const float*)d_in[7];
    const float* b_hh    = (const float*)d_in[8];
    const float* W_he    = (const float*)d_in[9];
    const float* b_he    = (const float*)d_in[10];
    const float* W_mu    = (const float*)d_in[11];
    const float* b_mu    = (const float*)d_in[12];
    const float* W_lv    = (const float*)d_in[13];
    const float* b_lv    = (const float*)d_in[14];
    const float* h_inf   = (const float*)d_in[15];
    const float* ti      = (const float*)d_in[16];
    const float* bi      = (const float*)d_in[17];
    (void)in_sizes; (void)n_in; (void)out_size; (void)ws_size;

    float* ws = (float*)d_ws;
    size_t off = 0;
    float* phi      = ws + off; off += (size_t)NROWS * E_DIM;                 // 64 MB
    float* xg       = ws + off; off += (size_t)NROWS * G3_DIM;                // 384 MB
    float* Hbuf     = ws + off; off += (size_t)(T_DIM + 1) * B_DIM * H_DIM;   // 128 MB
    float* he       = ws + off; off += (size_t)NROWS * S_DIM;                 // 32 MB
    float* partial  = ws + off; off += (size_t)NROWS;
    float* partial2 = ws + off; off += 256;

    // h0 = 0 (Hbuf slab 0); rest of Hbuf is written before read.
    hipMemsetAsync(Hbuf, 0, (size_t)B_DIM * H_DIM * sizeof(float), stream);

    // 1) input embedding (WMMA f32)
    k_embed<<<(NROWS / 16) * (E_DIM / 16) / 8, 256, 0, stream>>>(
        x, W_embed, b_embed, phi);

    // 2) all input-side gate GEMMs, hoisted out of the recurrence (WMMA f32)
    k_xgates<<<(NROWS / 16) * (G3_DIM / 16) / 8, 256, 0, stream>>>(
        phi, W_ih, b_ih, xg);

    // 3) sequential GRU scan: 512 shallow steps; W_hh strips staged in LDS
    //    via the Tensor Data Mover, B-fragments reused by 8 waves per block.
    for (int step = 0; step < T_DIM; ++step) {
        const float* Hin  = Hbuf + (size_t)step       * B_DIM * H_DIM;
        float*       Hout = Hbuf + (size_t)(step + 1) * B_DIM * H_DIM;
        const float* xgt  = xg   + (size_t)step       * B_DIM * G3_DIM;
        k_gru_step<<<H_DIM / 16, 256, GRU_SMEM_BYTES, stream>>>(
            Hin, Hout, xgt, W_hh, b_hh);
    }

    // 4) hidden-state embedding on the pre-update states hs = Hbuf[0..T-1]
    k_he<<<(NROWS / 16) * (S_DIM / 16) / 8, 256, 0, stream>>>(
        Hbuf, W_he, b_he, he);

    // 5) per-row Gaussian + RMTPP log-likelihoods -> partial[n]
    k_head<<<NROWS / 8, 256, 0, stream>>>(
        he, x, t, mask, W_mu, b_mu, W_lv, b_lv, h_inf, ti, bi, partial);

    // 6) deterministic reduction 65536 -> 256 -> 1
    k_reduce<<<256, 256, 0, stream>>>(partial, partial2);
    k_reduce<<<1, 256, 0, stream>>>(partial2, (float*)d_out);
}